// TokenEmbeddingPoolerWithAttentions_4415226380945
// MI455X (gfx1250) — compile-verified
//
#include <hip/hip_runtime.h>
#include <math.h>

// Sizes from the reference
#define BB 32
#define SS 2048
#define HH 768
#define OO 768
#define KK 64

typedef __attribute__((ext_vector_type(2))) float v2f;
typedef __attribute__((ext_vector_type(8))) float v8f;

__device__ __forceinline__ float waveReduceSum(float v) {
  // wave32 butterfly reduction
  #pragma unroll
  for (int off = 16; off > 0; off >>= 1)
    v += __shfl_xor(v, off, 32);
  return v;
}

// -------- Kernel 1: per-batch attention (gather only the 64 masked rows) ----
// grid = B blocks, block = 256 threads (8 wave32)
__global__ __launch_bounds__(256)
void attn_pool_kernel(const float* __restrict__ hidden,        // B*S*H
                      const int*   __restrict__ token_idxs,    // B*K
                      const float* __restrict__ subject_hidden,// B*H
                      const float* __restrict__ Wa,            // 2H
                      const float* __restrict__ ba,            // 1
                      float* __restrict__ out_attn,            // B*S (d_out + B*O)
                      float* __restrict__ pooled)              // B*H (workspace)
{
  __shared__ float sh_scores[KK];
  __shared__ int   sh_idx[KK];
  __shared__ float sh_w[KK];       // softmax weight, zeroed for duplicate entries
  __shared__ float sh_red[8];
  __shared__ float sh_max, sh_den;

  const int b    = blockIdx.x;
  const int t    = threadIdx.x;
  const int lane = t & 31;
  const int wid  = t >> 5;

  const float* subj = subject_hidden + (size_t)b * HH;
  const float* hid_b = hidden + (size_t)b * SS * HH;

  // ---- Phase A: partial dot(subject, Wa[0:H]) ----
  float ps = 0.f;
  #pragma unroll
  for (int j = t; j < HH; j += 256) ps += subj[j] * Wa[j];
  ps = waveReduceSum(ps);
  if (lane == 0) sh_red[wid] = ps;

  // ---- Phase B: per-token scores, 8 tokens per wave ----
  #pragma unroll
  for (int j = 0; j < KK / 8; ++j) {
    int k = wid + 8 * j;                 // token slot
    int idx = token_idxs[b * KK + k];
    const float* row = hid_b + (size_t)idx * HH;
    float acc = 0.f;
    #pragma unroll
    for (int c = lane; c < HH; c += 32) acc += row[c] * Wa[HH + c];
    acc = waveReduceSum(acc);
    if (lane == 0) { sh_scores[k] = acc; sh_idx[k] = idx; }
  }

  // ---- zero the full attention output row (S entries) ----
  #pragma unroll
  for (int i = t; i < SS; i += 256) out_attn[(size_t)b * SS + i] = 0.f;

  __syncthreads();

  // ---- Phase C (serial, tiny): add subj term, dedupe, max, denom ----
  if (t == 0) {
    float subjba = ba[0];
    #pragma unroll
    for (int w = 0; w < 8; ++w) subjba += sh_red[w];
    float m = -1e30f;
    #pragma unroll
    for (int k = 0; k < KK; ++k) {
      sh_scores[k] += subjba;
      m = fmaxf(m, sh_scores[k]);
    }
    float den = 0.f;
    #pragma unroll
    for (int k = 0; k < KK; ++k) {
      bool valid = (k == 0) || (sh_idx[k] != sh_idx[k - 1]); // sorted -> dedupe
      if (valid) den += expf(sh_scores[k] - m);
    }
    sh_max = m; sh_den = den;
  }
  __syncthreads();

  // ---- softmax weights: scatter to output, keep deduped copy for pooling ----
  if (t < KK) {
    bool valid = (t == 0) || (sh_idx[t] != sh_idx[t - 1]);
    float w = expf(sh_scores[t] - sh_max) / sh_den;
    out_attn[(size_t)b * SS + sh_idx[t]] = w;   // duplicates write same value
    sh_w[t] = valid ? w : 0.f;
  }
  __syncthreads();

  // ---- Phase D: pooled[b, :] = sum_k w_k * hidden[b, idx_k, :] ----
  float acc0 = 0.f, acc1 = 0.f, acc2 = 0.f;
  #pragma unroll 8
  for (int k = 0; k < KK; ++k) {
    float w = sh_w[k];
    const float* row = hid_b + (size_t)sh_idx[k] * HH;
    acc0 += w * row[t];
    acc1 += w * row[t + 256];
    acc2 += w * row[t + 512];
  }
  pooled[(size_t)b * HH + t]       = acc0;
  pooled[(size_t)b * HH + t + 256] = acc1;
  pooled[(size_t)b * HH + t + 512] = acc2;
}

// -------- Kernel 2: transformed = tanh(pooled @ Wo + bo) via f32 WMMA -------
// 32x768 @ 768x768. 2x48 = 96 tiles of 16x16, one wave per tile.
// grid = 24 blocks, block = 128 threads (4 wave32)
__global__ __launch_bounds__(128)
void gemm_tanh_kernel(const float* __restrict__ A,   // pooled 32x768
                      const float* __restrict__ Wo,  // 768x768
                      const float* __restrict__ bo,  // 768
                      float* __restrict__ out)       // 32x768
{
  const int lane = threadIdx.x & 31;
  const int wid  = threadIdx.x >> 5;
  const int tile = blockIdx.x * 4 + wid;   // 0..95
  const int rowBase = (tile / 48) * 16;    // 0 or 16
  const int colBase = (tile % 48) * 16;

  // ISA 16x4 f32 A layout: lanes 0-15 -> K=0,1 ; lanes 16-31 -> K=2,3
  const int mn   = lane & 15;          // M for A, N for B/C/D
  const int koff = (lane >> 4) * 2;    // 0 or 2

  const float* arow = A  + (size_t)(rowBase + mn) * HH + koff;        // A[m][koff + k0]
  const float* bcol = Wo + (size_t)koff * OO + colBase + mn;          // Wo[koff + k0][n]

  v8f c = {};
  #pragma unroll 4
  for (int k0 = 0; k0 < HH; k0 += 4) {
    v2f a, b;
    a.x = arow[k0];
    a.y = arow[k0 + 1];
    b.x = bcol[(size_t)k0 * OO];
    b.y = bcol[(size_t)(k0 + 1) * OO];
    // D = A(16x4) * B(4x16) + C, full f32
    c = __builtin_amdgcn_wmma_f32_16x16x4_f32(
        /*neg_a=*/false, a, /*neg_b=*/false, b,
        /*c_mod=*/(short)0, c, /*reuse_a=*/false, /*reuse_b=*/false);
  }

  // C/D layout: VGPR r -> M = r + (lane>>4)*8, N = lane&15
  const int mrow = (lane >> 4) * 8;
  const float bias = bo[colBase + mn];
  #pragma unroll
  for (int r = 0; r < 8; ++r) {
    out[(size_t)(rowBase + mrow + r) * OO + colBase + mn] = tanhf(c[r] + bias);
  }
}

extern "C" void kernel_launch(void* const* d_in, const int* in_sizes, int n_in,
                              void* d_out, int out_size, void* d_ws, size_t ws_size,
                              hipStream_t stream) {
  const float* hidden         = (const float*)d_in[0]; // B*S*H
  const int*   token_idxs     = (const int*)  d_in[1]; // B*K
  const float* subject_hidden = (const float*)d_in[2]; // B*H
  const float* Wa             = (const float*)d_in[3]; // 2H
  const float* ba             = (const float*)d_in[4]; // 1
  const float* Wo             = (const float*)d_in[5]; // H*O
  const float* bo             = (const float*)d_in[6]; // O

  float* out_transformed = (float*)d_out;                    // B*O
  float* out_attn        = (float*)d_out + (size_t)BB * OO;  // B*S
  float* pooled          = (float*)d_ws;                     // B*H scratch

  attn_pool_kernel<<<BB, 256, 0, stream>>>(hidden, token_idxs, subject_hidden,
                                           Wa, ba, out_attn, pooled);
  gemm_tanh_kernel<<<24, 128, 0, stream>>>(pooled, Wo, bo, out_transformed);
}